// SocialSTGCNModel_51479478010515
// MI455X (gfx1250) — compile-verified
//
#include <hip/hip_runtime.h>
#include <hip/hip_bf16.h>

typedef __attribute__((ext_vector_type(2))) float v2f;
typedef __attribute__((ext_vector_type(8))) float v8f;

#define T_OBS 16
#define NAG   4096
#define NV    8
#define DYN   64
#define ENC   64
#define DEC   128
#define PRED  25
#define GH    13
#define GW    3
#define SOCD  64
#define C31   16
#define SOC_EMB 80   // ((13-4+1)//2)*16
#define XCDIM 112    // SELF(32) + SOC_EMB(80)
#define NWAVES 4     // cooperative waves per recurrent block

// ---------- WMMA helpers (fp32 16x16x4) ----------

__device__ __forceinline__ v8f wmma4(v2f a, v2f b, v8f c) {
  // 8 args: (neg_a, A, neg_b, B, c_mod, C, reuse_a, reuse_b)
  return __builtin_amdgcn_wmma_f32_16x16x4_f32(false, a, false, b, (short)0, c, false, false);
}

// A fragment: 16x4 tile from row-major A (rows 0..15), columns k0..k0+3
__device__ __forceinline__ v2f a_frag(const float* A, int lda, int k0, int lane) {
  int row = lane & 15;
  int kb  = (lane >> 4) << 1;          // 0 or 2
  const float* p = A + row * lda + k0 + kb;
  v2f a; a.x = p[0]; a.y = p[1];
  return a;
}

// B fragment: 4x16 tile of B = W^T, where W is row-major (Nrows, K).
// B[k][n] = W[n][k];  tile columns n0..n0+15, k0..k0+3
__device__ __forceinline__ v2f b_fragT(const float* W, int ldw, int n0, int k0, int lane) {
  int col = lane & 15;
  int kb  = (lane >> 4) << 1;
  const float* p = W + (size_t)(n0 + col) * ldw + k0 + kb;
  v2f b; b.x = p[0]; b.y = p[1];
  return b;
}

__device__ __forceinline__ float leaky(float v) { return v >= 0.f ? v : 0.1f * v; }
__device__ __forceinline__ float sigm(float x)  { return 1.0f / (1.0f + __expf(-x)); }

// Cooperative GEMM: C = A(16xK) * W^T(KxNcols) [+bias] [leaky].
// The Ncols/16 output tiles are strided across `nw` waves (wave id `wv`).
__device__ __forceinline__ void gemm_rowtile(const float* A, int lda,
                                             const float* W, int ldw, int Kdim,
                                             const float* bias,
                                             float* out, int ldo, int Ncols,
                                             int lane, int wv, int nw, int act) {
  for (int n0 = wv * 16; n0 < Ncols; n0 += 16 * nw) {
    v8f acc = {};
    for (int k0 = 0; k0 < Kdim; k0 += 4) {
      v2f a = a_frag(A, lda, k0, lane);
      v2f b = b_fragT(W, ldw, n0, k0, lane);
      acc = wmma4(a, b, acc);
    }
    int col   = lane & 15;
    int rbase = (lane >> 4) << 3;      // 0 or 8
    float bv = bias ? bias[n0 + col] : 0.0f;
#pragma unroll
    for (int r = 0; r < 8; ++r) {
      float v = acc[r] + bv;
      if (act) v = leaky(v);
      out[(size_t)(rbase + r) * ldo + n0 + col] = v;
    }
  }
}

// GRU gate update over 16 rows, hidden H; gi/gh ld = 3H, h ld = H (all LDS)
__device__ __forceinline__ void gru_update(const float* gi, const float* gh,
                                           float* h, int H, int tid, int nthr) {
  int total = 16 * H;
  for (int e = tid; e < total; e += nthr) {
    int row = e / H, c = e - row * H;
    const float* gir = gi + (size_t)row * 3 * H;
    const float* ghr = gh + (size_t)row * 3 * H;
    float r = sigm(gir[c]         + ghr[c]);
    float z = sigm(gir[H + c]     + ghr[H + c]);
    float n = tanhf(gir[2*H + c]  + r * ghr[2*H + c]);
    h[(size_t)row * H + c] = (1.f - z) * n + z * h[(size_t)row * H + c];
  }
}

// ---------- Kernel 1: ego encoder GRU (fused over T, 4 cooperative waves) ----------
__global__ __launch_bounds__(32 * NWAVES)
void enc_x_kernel(const float* __restrict__ x, const float* __restrict__ dyn_W,
                  const float* __restrict__ dyn_b,
                  const float* __restrict__ Wih, const float* __restrict__ Whh,
                  const float* __restrict__ bih, const float* __restrict__ bhh,
                  float* __restrict__ h_x) {
  __shared__ float xe[16 * 64];
  __shared__ float gi[16 * 192];
  __shared__ float gh[16 * 192];
  __shared__ float h [16 * 64];
  int tid = threadIdx.x, lane = tid & 31, wv = tid >> 5;
  const int nthr = 32 * NWAVES;
  int n0 = blockIdx.x * 16;
  for (int e = tid; e < 16 * 64; e += nthr) h[e] = 0.f;
  __syncthreads();
  for (int t = 0; t < T_OBS; ++t) {
    // dyn embed (K=2, VALU): xe = leaky(x_t @ dyn_W^T + dyn_b)
    for (int e = tid; e < 16 * 64; e += nthr) {
      int row = e >> 6, c = e & 63;
      const float* xp = x + ((size_t)t * NAG + n0 + row) * 2;
      xe[e] = leaky(xp[0] * dyn_W[c * 2] + xp[1] * dyn_W[c * 2 + 1] + dyn_b[c]);
    }
    __syncthreads();
    gemm_rowtile(xe, 64, Wih, 64, 64, bih, gi, 192, 192, lane, wv, NWAVES, 0);
    gemm_rowtile(h,  64, Whh, 64, 64, bhh, gh, 192, 192, lane, wv, NWAVES, 0);
    __syncthreads();
    gru_update(gi, gh, h, 64, tid, nthr);
    __syncthreads();
  }
  for (int e = tid; e < 16 * 64; e += nthr) h_x[(size_t)n0 * 64 + e] = h[e];
}

// ---------- Kernel 2: neighbor pipeline + encoder GRU (fully fused) ----------
__global__ __launch_bounds__(32 * NWAVES)
void enc_ng_kernel(const float* __restrict__ ngbrs, const float* __restrict__ graphs,
                   const float* __restrict__ dyn_W, const float* __restrict__ dyn_b,
                   const float* __restrict__ s_W, const float* __restrict__ s_b,
                   const float* __restrict__ Wih, const float* __restrict__ Whh,
                   const float* __restrict__ bih, const float* __restrict__ bhh,
                   float* __restrict__ h_ng) {
  __shared__ float nge[16 * 64];
  __shared__ float hs [16 * 128];
  __shared__ float agg[16 * 64];
  __shared__ float gi [16 * 192];
  __shared__ float gh [16 * 192];
  __shared__ float h  [16 * 64];
  int tid = threadIdx.x, lane = tid & 31, wv = tid >> 5;
  const int nthr = 32 * NWAVES;
  int row0 = blockIdx.x * 16;          // batch rows over N*V = 32768 (2 agents)
  int nA = row0 >> 3;                  // first agent id
  for (int e = tid; e < 16 * 64; e += nthr) h[e] = 0.f;
  __syncthreads();
  for (int t = 0; t < T_OBS; ++t) {
    // dyn embed of neighbors
    for (int e = tid; e < 16 * 64; e += nthr) {
      int row = e >> 6, c = e & 63;
      const float* xp = ngbrs + ((size_t)t * (NAG * NV) + row0 + row) * 2;
      nge[e] = leaky(xp[0] * dyn_W[c * 2] + xp[1] * dyn_W[c * 2 + 1] + dyn_b[c]);
    }
    __syncthreads();
    // hs = nge @ s_W^T + s_b  : (16 x 128)
    gemm_rowtile(nge, 64, s_W, 64, 64, s_b, hs, 128, 128, lane, wv, NWAVES, 0);
    __syncthreads();
    // agg[a*8+w, c] = sum_{v,k} hs[a*8+v, k*64+c] * graphs[nA+a, k, v, w]
    for (int e = tid; e < 16 * 64; e += nthr) {
      int row = e >> 6, c = e & 63;
      int a = row >> 3, w = row & 7;
      const float* g = graphs + (size_t)(nA + a) * 128;   // (K=2, V=8, W=8)
      float accv = 0.f;
#pragma unroll
      for (int k = 0; k < 2; ++k)
#pragma unroll
        for (int v = 0; v < 8; ++v)
          accv += hs[(a * 8 + v) * 128 + k * 64 + c] * g[(k * 8 + v) * 8 + w];
      agg[e] = accv;
    }
    __syncthreads();
    gemm_rowtile(agg, 64, Wih, 64, 64, bih, gi, 192, 192, lane, wv, NWAVES, 0);
    gemm_rowtile(h,   64, Whh, 64, 64, bhh, gh, 192, 192, lane, wv, NWAVES, 0);
    __syncthreads();
    gru_update(gi, gh, h, 64, tid, nthr);
    __syncthreads();
  }
  for (int e = tid; e < 16 * 64; e += nthr) h_ng[(size_t)row0 * 64 + e] = h[e];
}

// ---------- Kernel 3: masked scatter into social grid (N, ENC, GH, GW) ----------
__global__ __launch_bounds__(64)
void scatter_kernel(const unsigned char* __restrict__ masks,
                    const float* __restrict__ h_ng,
                    float* __restrict__ soc_t) {
  __shared__ int rank[GW * GH];
  int n = blockIdx.x;
  if (threadIdx.x == 0) {
    int r = 0;
    for (int j = 0; j < GW * GH; ++j) {
      bool m = masks[((size_t)n * (GW * GH) + j) * ENC] != 0;
      rank[j] = m ? r++ : -1;
    }
  }
  __syncthreads();
  int c = threadIdx.x;                 // channel 0..63
  for (int j = 0; j < GW * GH; ++j) {
    int gw = j / GH, gh = j - gw * GH;
    float v = 0.f;
    int rk = rank[j];
    if (rk >= 0) v = h_ng[((size_t)n * NV + rk) * ENC + c];
    soc_t[(((size_t)n * ENC + c) * GH + gh) * GW + gw] = v;
  }
}

// ---------- Kernel 4: 3x3 conv 64->64 VALID on (13,3) -> (11,1), leaky ----------
__global__ __launch_bounds__(32)
void conv1_kernel(const float* __restrict__ soc_t, const float* __restrict__ soc_W,
                  const float* __restrict__ soc_b, float* __restrict__ c1) {
  int lane = threadIdx.x;
  int n = blockIdx.x;
  v8f acc[4] = {v8f{}, v8f{}, v8f{}, v8f{}};
  const float* in = soc_t + (size_t)n * ENC * GH * GW;
  int y  = lane & 15;
  int kb = (lane >> 4) << 1;
  for (int ky = 0; ky < 3; ++ky)
    for (int kx = 0; kx < 3; ++kx)
      for (int k0 = 0; k0 < 64; k0 += 4) {
        int ic = k0 + kb;
        v2f a; a.x = 0.f; a.y = 0.f;
        if (y < 11) {
          a.x = in[((size_t)ic       * GH + y + ky) * GW + kx];
          a.y = in[((size_t)(ic + 1) * GH + y + ky) * GW + kx];
        }
#pragma unroll
        for (int nt = 0; nt < 4; ++nt) {
          int oc = nt * 16 + (lane & 15);
          v2f b;
          b.x = soc_W[(((size_t)oc * 64 + ic)     * 3 + ky) * 3 + kx];
          b.y = soc_W[(((size_t)oc * 64 + ic + 1) * 3 + ky) * 3 + kx];
          acc[nt] = wmma4(a, b, acc[nt]);
        }
      }
  int col = lane & 15;
  int rbase = (lane >> 4) << 3;
#pragma unroll
  for (int nt = 0; nt < 4; ++nt) {
    int oc = nt * 16 + col;
    float bv = soc_b[oc];
#pragma unroll
    for (int r = 0; r < 8; ++r) {
      int row = rbase + r;
      if (row < 11)
        c1[((size_t)n * SOCD + oc) * 11 + row] = leaky(acc[nt][r] + bv);
    }
  }
}

// ---------- Kernel 5: 3x1 conv 64->16 -> (9,1), leaky, maxpool -> emb(N,80) ----------
__global__ __launch_bounds__(32)
void conv2_kernel(const float* __restrict__ c1, const float* __restrict__ c31_W,
                  const float* __restrict__ c31_b, float* __restrict__ emb) {
  __shared__ float tile[16 * 16];      // [y][oc]
  int lane = threadIdx.x;
  int n = blockIdx.x;
  v8f acc = {};
  const float* in = c1 + (size_t)n * SOCD * 11;
  int y  = lane & 15;
  int kb = (lane >> 4) << 1;
  for (int ky = 0; ky < 3; ++ky)
    for (int k0 = 0; k0 < 64; k0 += 4) {
      int ic = k0 + kb;
      v2f a; a.x = 0.f; a.y = 0.f;
      if (y < 9) {
        a.x = in[(size_t)ic       * 11 + y + ky];
        a.y = in[(size_t)(ic + 1) * 11 + y + ky];
      }
      int oc = lane & 15;
      v2f b;
      b.x = c31_W[((size_t)oc * 64 + ic)     * 3 + ky];
      b.y = c31_W[((size_t)oc * 64 + ic + 1) * 3 + ky];
      acc = wmma4(a, b, acc);
    }
  int col = lane & 15;
  int rbase = (lane >> 4) << 3;
  float bv = c31_b[col];
#pragma unroll
  for (int r = 0; r < 8; ++r)
    tile[(rbase + r) * 16 + col] = leaky(acc[r] + bv);
  __syncthreads();
  if (lane < 16) {
    int oc = lane;
    float* out = emb + (size_t)n * SOC_EMB + oc * 5;
    out[0] = tile[0 * 16 + oc];        // pad -inf window: just in[0]
#pragma unroll
    for (int p = 1; p < 5; ++p)
      out[p] = fmaxf(tile[(2 * p - 1) * 16 + oc], tile[(2 * p) * 16 + oc]);
  }
}

// ---------- Kernel 6: decoder GRU (gi computed once), 4 cooperative waves ----------
__global__ __launch_bounds__(32 * NWAVES)
void dec_kernel(const float* __restrict__ h_x, const float* __restrict__ hid_W,
                const float* __restrict__ hid_b, const float* __restrict__ emb,
                const float* __restrict__ Wih, const float* __restrict__ Whh,
                const float* __restrict__ bih, const float* __restrict__ bhh,
                const float* __restrict__ out_W, const float* __restrict__ out_b,
                float* __restrict__ out) {
  __shared__ float xc[16 * XCDIM];
  __shared__ float gi[16 * 384];
  __shared__ float gh[16 * 384];
  __shared__ float h [16 * 128];
  int tid = threadIdx.x, lane = tid & 31, wv = tid >> 5;
  const int nthr = 32 * NWAVES;
  int n0 = blockIdx.x * 16;
  // x_s = leaky(h_x @ hid_W^T + hid_b) into xc[:, 0:32]  (2 tiles: waves 0,1)
  gemm_rowtile(h_x + (size_t)n0 * 64, 64, hid_W, 64, 64, hid_b, xc, XCDIM, 32,
               lane, wv, NWAVES, 1);
  // soc embedding into xc[:, 32:112]
  for (int e = tid; e < 16 * SOC_EMB; e += nthr) {
    int row = e / SOC_EMB, c = e - row * SOC_EMB;
    xc[row * XCDIM + 32 + c] = emb[(size_t)(n0 + row) * SOC_EMB + c];
  }
  for (int e = tid; e < 16 * 128; e += nthr) h[e] = 0.f;
  __syncthreads();
  // gi constant over time: (16 x 384)
  gemm_rowtile(xc, XCDIM, Wih, XCDIM, XCDIM, bih, gi, 384, 384, lane, wv, NWAVES, 0);
  __syncthreads();
  for (int t = 0; t < PRED; ++t) {
    gemm_rowtile(h, 128, Whh, 128, 128, bhh, gh, 384, 384, lane, wv, NWAVES, 0);
    __syncthreads();
    gru_update(gi, gh, h, 128, tid, nthr);
    __syncthreads();
    // o = h @ out_W^T + out_b, then mu / exp(sig) / tanh(rho)
    for (int e = tid; e < 16 * 5; e += nthr) {
      int row = e / 5, j = e - row * 5;
      const float* hr = h + row * 128;
      const float* w  = out_W + j * 128;
      float a = out_b[j];
      for (int c = 0; c < 128; ++c) a += hr[c] * w[c];
      if (j == 2 || j == 3) a = __expf(a);
      else if (j == 4)      a = tanhf(a);
      out[((size_t)t * NAG + n0 + row) * 5 + j] = a;
    }
    __syncthreads();
  }
}

// ---------- launch ----------
extern "C" void kernel_launch(void* const* d_in, const int* in_sizes, int n_in,
                              void* d_out, int out_size, void* d_ws, size_t ws_size,
                              hipStream_t stream) {
  (void)in_sizes; (void)n_in; (void)out_size; (void)ws_size;
  const float* x        = (const float*)d_in[0];
  const float* ngbrs    = (const float*)d_in[1];
  const float* graphs   = (const float*)d_in[2];
  const unsigned char* masks = (const unsigned char*)d_in[3];
  // d_in[4] = ngbrs_idx (unused: it is arange(N+1)*V)
  const float* dyn_W    = (const float*)d_in[5];
  const float* dyn_b    = (const float*)d_in[6];
  const float* enc_Wih  = (const float*)d_in[7];
  const float* enc_Whh  = (const float*)d_in[8];
  const float* enc_bih  = (const float*)d_in[9];
  const float* enc_bhh  = (const float*)d_in[10];
  const float* s_W      = (const float*)d_in[11];
  const float* s_b      = (const float*)d_in[12];
  const float* hid_W    = (const float*)d_in[13];
  const float* hid_b    = (const float*)d_in[14];
  const float* soc_W    = (const float*)d_in[15];
  const float* soc_b    = (const float*)d_in[16];
  const float* c31_W    = (const float*)d_in[17];
  const float* c31_b    = (const float*)d_in[18];
  const float* dec_Wih  = (const float*)d_in[19];
  const float* dec_Whh  = (const float*)d_in[20];
  const float* dec_bih  = (const float*)d_in[21];
  const float* dec_bhh  = (const float*)d_in[22];
  const float* out_W    = (const float*)d_in[23];
  const float* out_b    = (const float*)d_in[24];
  float* out = (float*)d_out;

  float* ws    = (float*)d_ws;
  float* h_x   = ws;                                   // 4096*64
  float* h_ng  = h_x  + (size_t)NAG * ENC;             // 32768*64
  float* soc_t = h_ng + (size_t)NAG * NV * ENC;        // 4096*64*13*3
  float* c1    = soc_t + (size_t)NAG * ENC * GH * GW;  // 4096*64*11
  float* emb   = c1   + (size_t)NAG * SOCD * 11;       // 4096*80

  hipLaunchKernelGGL(enc_x_kernel, dim3(NAG / 16), dim3(32 * NWAVES), 0, stream,
                     x, dyn_W, dyn_b, enc_Wih, enc_Whh, enc_bih, enc_bhh, h_x);
  hipLaunchKernelGGL(enc_ng_kernel, dim3(NAG * NV / 16), dim3(32 * NWAVES), 0, stream,
                     ngbrs, graphs, dyn_W, dyn_b, s_W, s_b,
                     enc_Wih, enc_Whh, enc_bih, enc_bhh, h_ng);
  hipLaunchKernelGGL(scatter_kernel, dim3(NAG), dim3(64), 0, stream,
                     masks, h_ng, soc_t);
  hipLaunchKernelGGL(conv1_kernel, dim3(NAG), dim3(32), 0, stream,
                     soc_t, soc_W, soc_b, c1);
  hipLaunchKernelGGL(conv2_kernel, dim3(NAG), dim3(32), 0, stream,
                     c1, c31_W, c31_b, emb);
  hipLaunchKernelGGL(dec_kernel, dim3(NAG / 16), dim3(32 * NWAVES), 0, stream,
                     h_x, hid_W, hid_b, emb, dec_Wih, dec_Whh, dec_bih, dec_bhh,
                     out_W, out_b, out);
}